// WeightedAStep_24524263260340
// MI455X (gfx1250) — compile-verified
//
#include <hip/hip_runtime.h>

// Problem constants (from reference): N=32768, D=4096, K=8, ridge=1e-6.
#define NROWS 32768
#define DDIM  4096
#define KDIM  8
#define RIDGE 1e-6f

typedef float v2f __attribute__((ext_vector_type(2)));
typedef float v8f __attribute__((ext_vector_type(8)));

// One wave owns a 16-row tile of n and reduces over all of D with
// V_WMMA_F32_16X16X4_F32:
//   P  = W      @ H   where H[d, 16j+c] = G[d, 2j + c/8] * G[d, c%8]   (4 tiles)
//   b  = (W∘X)  @ G                                                    (1 tile)
// Then a per-row 8x8 SPD solve in the epilogue.
__global__ __launch_bounds__(256) void wls_fused_kernel(
    const float* __restrict__ X, const float* __restrict__ W,
    const float* __restrict__ G, float* __restrict__ out)
{
    __shared__ float smem[128 * 72];   // 8 waves * 16 rows * (64 M + 8 b)

    const int lane = threadIdx.x & 31;
    const int wave = threadIdx.x >> 5;
    const int c    = lane & 15;        // B/C/D column index (kl within tile)
    const int h    = lane >> 4;        // lane half -> which pair of d's / row half
    const int l    = c & 7;            // l index of (k,l)
    const int kh   = c >> 3;           // k parity within a k-pair
    const int nbase = blockIdx.x * 128 + wave * 16;
    const int nl    = nbase + c;       // A-matrix row (n) this lane feeds

    const float* wrow = W + (size_t)nl * DDIM;
    const float* xrow = X + (size_t)nl * DDIM;

    v8f accM[4];
    v8f accB = (v8f)(0.0f);
#pragma unroll
    for (int j = 0; j < 4; ++j) accM[j] = (v8f)(0.0f);

#pragma unroll 2
    for (int d0 = 0; d0 < DDIM; d0 += 4) {
        const int dl = d0 + 2 * h;                 // this lane-half's d pair
        v2f wv = *(const v2f*)(wrow + dl);         // A fragment: W[n, dl..dl+1]
        v2f xv = *(const v2f*)(xrow + dl);

        const float* g0 = G + dl * KDIM;           // G rows dl, dl+1 (L2-hot)
        const float g0l = g0[l];
        const float g1l = g0[KDIM + l];

        // b-GEMM: A' = W∘X, B' cols 0..7 = G (cols 8..15 garbage, ignored)
        v2f awx; awx.x = wv.x * xv.x; awx.y = wv.y * xv.y;
        v2f bb;  bb.x  = g0l;         bb.y  = g1l;
        accB = __builtin_amdgcn_wmma_f32_16x16x4_f32(
            false, awx, false, bb, (short)0, accB, false, false);

        // M-GEMM: 4 column tiles of H[d, kl] = G[d,k]*G[d,l], shared A = W tile
#pragma unroll
        for (int j = 0; j < 4; ++j) {
            const float gk0 = g0[2 * j + kh];
            const float gk1 = g0[KDIM + 2 * j + kh];
            v2f bj; bj.x = gk0 * g0l; bj.y = gk1 * g1l;
            accM[j] = __builtin_amdgcn_wmma_f32_16x16x4_f32(
                false, wv, false, bj, (short)0, accM[j], false, false);
        }
    }

    // ---- Epilogue: spill accumulators to per-wave LDS slab, re-own per row ----
    float* rowbuf = smem + wave * (16 * 72);
#pragma unroll
    for (int j = 0; j < 4; ++j) {
#pragma unroll
        for (int r = 0; r < 8; ++r) {
            // D[M=r+8h, N=c] of tile j -> M_n[2j + c/8, c%8], n = nbase + r + 8h
            rowbuf[(r + 8 * h) * 72 + (2 * j + kh) * 8 + l] = accM[j][r];
        }
    }
    if (c < 8) {
#pragma unroll
        for (int r = 0; r < 8; ++r)
            rowbuf[(r + 8 * h) * 72 + 64 + c] = accB[r];
    }

    // Lanes 0..15: one 8x8 SPD solve each (same-wave LDS ops stay ordered).
    if (lane < 16) {
        const float* mb = rowbuf + lane * 72;
        float m[8][8], bv[8];
#pragma unroll
        for (int i = 0; i < 8; ++i) {
#pragma unroll
            for (int jj = 0; jj < 8; ++jj) m[i][jj] = mb[i * 8 + jj];
            m[i][i] += RIDGE;
            bv[i] = mb[64 + i];
        }
        // Gaussian elimination, no pivoting (M is SPD + ridge)
#pragma unroll
        for (int p = 0; p < 8; ++p) {
            const float inv = 1.0f / m[p][p];
#pragma unroll
            for (int col = 0; col < 8; ++col)
                if (col > p) m[p][col] *= inv;
            bv[p] *= inv;
#pragma unroll
            for (int row = 0; row < 8; ++row) {
                if (row > p) {
                    const float f = m[row][p];
#pragma unroll
                    for (int col = 0; col < 8; ++col)
                        if (col > p) m[row][col] -= f * m[p][col];
                    bv[row] -= f * bv[p];
                }
            }
        }
        float a[8];
#pragma unroll
        for (int p = 7; p >= 0; --p) {
            float s = bv[p];
#pragma unroll
            for (int col = 0; col < 8; ++col)
                if (col > p) s -= m[p][col] * a[col];
            a[p] = s;                    // diagonal normalized to 1
        }
        float* orow = out + (size_t)(nbase + lane) * KDIM;
#pragma unroll
        for (int k = 0; k < 8; ++k) orow[k] = a[k];
    }
}

extern "C" void kernel_launch(void* const* d_in, const int* in_sizes, int n_in,
                              void* d_out, int out_size, void* d_ws, size_t ws_size,
                              hipStream_t stream) {
    const float* X = (const float*)d_in[0];
    const float* W = (const float*)d_in[1];
    const float* G = (const float*)d_in[2];
    float* out = (float*)d_out;

    dim3 block(256);                       // 8 wave32s; each wave owns 16 rows
    dim3 grid(NROWS / 128);                // 256 workgroups
    hipLaunchKernelGGL(wls_fused_kernel, grid, block, 0, stream, X, W, G, out);
}